// GCN2_GAT_89008902243179
// MI455X (gfx1250) — compile-verified
//
#include <hip/hip_runtime.h>
#include <hip/hip_bf16.h>
#include <cstdint>

typedef __attribute__((ext_vector_type(2))) float v2f;
typedef __attribute__((ext_vector_type(8))) float v8f;

// ---------------- zero fill ----------------
__global__ void k_zero(float* __restrict__ p, size_t n) {
  size_t i = (size_t)blockIdx.x * blockDim.x + threadIdx.x;
  if (i < n) p[i] = 0.0f;
}

// ---------------- degree accumulation (edges + self loops) ----------------
__global__ void k_degree(const long long* __restrict__ ei,
                         const float* __restrict__ ew,
                         float* __restrict__ deg, int E, int N) {
  int i = blockIdx.x * blockDim.x + threadIdx.x;
  if (i >= E + N) return;
  long long d; float w;
  if (i < E) { d = ei[(size_t)E + i]; w = ew[i]; }   // edge_index[1][i]
  else       { d = i - E;             w = 1.0f;  }   // self loop
  atomicAdd(&deg[d], w);
}

__global__ void k_dinv(float* __restrict__ deg, int N) {
  int i = blockIdx.x * blockDim.x + threadIdx.x;
  if (i < N) {
    float v = deg[i];
    deg[i] = (v > 0.0f) ? rsqrtf(v) : 0.0f;
  }
}

// ---------------- f32 WMMA GEMM: C[M,Ncols] = A[M,K] @ B[K,Ncols] ----------------
// One wave per 16x16 output tile; K stepped by 4 via V_WMMA_F32_16X16X4_F32.
// Requires M % 16 == 0, Ncols % 16 == 0, K % 4 == 0 (true for this problem).
// ISA layouts (CDNA5 ISA):
//   A 16x4 f32 : lane m (0..15) row M=m; VGPR j holds K = j + 2*(lane>=16)
//   B 4x16 f32 : lane m col N=m;         VGPR j holds K = j + 2*(lane>=16)
//   C/D 16x16  : VGPR j -> row M = j + 8*(lane>=16), col N = lane&15
__global__ void k_gemm_wmma_f32(const float* __restrict__ A,
                                const float* __restrict__ B,
                                float* __restrict__ C,
                                int M, int Ncols, int K) {
  const int wavesPerBlock = blockDim.x >> 5;
  const int wave = blockIdx.x * wavesPerBlock + ((int)threadIdx.x >> 5);
  const int tilesN = Ncols >> 4;
  const int tilesM = M >> 4;
  if (wave >= tilesM * tilesN) return;
  const int tm   = wave / tilesN;
  const int tn   = wave - tm * tilesN;
  const int lane = (int)threadIdx.x & 31;
  const int half = lane >> 4;   // 0: lanes 0-15, 1: lanes 16-31
  const int m    = lane & 15;

  const float* arow = A + (size_t)(tm * 16 + m) * K;   // row of A for this lane
  const float* bcol = B + (tn * 16 + m);               // column of B for this lane

  v8f acc = {};
#pragma unroll 8
  for (int k0 = 0; k0 < K; k0 += 4) {
    const int ka = k0 + 2 * half;
    v2f a = *reinterpret_cast<const v2f*>(arow + ka);  // contiguous K pair, 8B aligned
    v2f b;
    b.x = bcol[(size_t)ka * Ncols];
    b.y = bcol[(size_t)(ka + 1) * Ncols];
    acc = __builtin_amdgcn_wmma_f32_16x16x4_f32(
        /*neg_a=*/false, a, /*neg_b=*/false, b,
        /*c_mod=*/(short)0, acc, /*reuse_a=*/false, /*reuse_b=*/false);
  }

  float* crow = C + (size_t)(tm * 16 + half * 8) * Ncols + tn * 16 + m;
#pragma unroll
  for (int j = 0; j < 8; ++j)
    crow[(size_t)j * Ncols] = acc[j];
}

// ---------------- normalized edge aggregation (one block per edge, C lanes) ----
__global__ void k_aggregate(const long long* __restrict__ ei,
                            const float* __restrict__ ew,
                            const float* __restrict__ dinv,
                            const float* __restrict__ hsrc,
                            float* __restrict__ hdst,
                            int E, int C) {
  const int e = blockIdx.x;
  long long s, d; float w;
  if (e < E) { s = ei[e]; d = ei[(size_t)E + e]; w = ew[e]; }
  else       { s = e - E; d = s;                 w = 1.0f;  }
  const float norm = dinv[s] * w * dinv[d];
  const int c = threadIdx.x;
  atomicAdd(&hdst[(size_t)d * C + c], norm * hsrc[(size_t)s * C + c]);
}

// ---------------- bias + relu (C is a power of two) ----------------
__global__ void k_bias_relu(float* __restrict__ h, const float* __restrict__ b,
                            size_t n, int cmask) {
  size_t i = (size_t)blockIdx.x * blockDim.x + threadIdx.x;
  if (i < n) h[i] = fmaxf(h[i] + b[i & (size_t)cmask], 0.0f);
}

// ---------------- bias + softmax over 64 columns, one wave32 per row --------
__global__ void k_bias_softmax64(float* __restrict__ out, const float* __restrict__ b,
                                 int N) {
  const int wavesPerBlock = blockDim.x >> 5;
  const int row = blockIdx.x * wavesPerBlock + ((int)threadIdx.x >> 5);
  if (row >= N) return;
  const int lane = (int)threadIdx.x & 31;
  float* p = out + (size_t)row * 64;
  float v0 = p[lane]      + b[lane];
  float v1 = p[lane + 32] + b[lane + 32];
  float mx = fmaxf(v0, v1);
#pragma unroll
  for (int off = 16; off > 0; off >>= 1) mx = fmaxf(mx, __shfl_xor(mx, off, 32));
  const float e0 = expf(v0 - mx);
  const float e1 = expf(v1 - mx);
  float s = e0 + e1;
#pragma unroll
  for (int off = 16; off > 0; off >>= 1) s += __shfl_xor(s, off, 32);
  const float inv = 1.0f / s;
  p[lane]      = e0 * inv;
  p[lane + 32] = e1 * inv;
}

extern "C" void kernel_launch(void* const* d_in, const int* in_sizes, int n_in,
                              void* d_out, int out_size, void* d_ws, size_t ws_size,
                              hipStream_t stream) {
  // Input order: x, edge_index, edge_weight, W_gat, att_src, att_dst, b_gat,
  //              W1, b1, W2, b2.  GAT weights are dead code in the reference.
  const float*     x  = (const float*)d_in[0];
  const long long* ei = (const long long*)d_in[1];   // int64 [2, E]
  const float*     ew = (const float*)d_in[2];
  const float*     W1 = (const float*)d_in[7];
  const float*     b1 = (const float*)d_in[8];
  const float*     W2 = (const float*)d_in[9];
  const float*     b2 = (const float*)d_in[10];

  const int FH = in_sizes[8];              // 256
  const int FO = in_sizes[10];             // 64
  const int FI = in_sizes[7] / FH;         // 128
  const int N  = in_sizes[0] / FI;         // 20000
  const int E  = in_sizes[1] / 2;          // 320000
  const int ET = E + N;                    // edges incl. self loops

  // Workspace layout (floats): [ dinv:N | h1pre:N*FH | h1:N*FH | h2pre:N*FO ]
  float* deg   = (float*)d_ws;             // becomes dinv in place
  float* h1pre = deg   + N;
  float* h1    = h1pre + (size_t)N * FH;
  float* h2pre = h1    + (size_t)N * FH;
  float* out   = (float*)d_out;
  (void)n_in; (void)out_size; (void)ws_size; (void)h2pre;

  const int TB = 256;
  auto blocks = [](size_t n, int tb) { return (unsigned)((n + tb - 1) / tb); };

  // 1) zero accumulation targets (harness poisons d_out/d_ws with 0xAA)
  k_zero<<<blocks((size_t)N, TB), TB, 0, stream>>>(deg, (size_t)N);
  k_zero<<<blocks((size_t)N * FH, TB), TB, 0, stream>>>(h1, (size_t)N * FH);
  k_zero<<<blocks((size_t)N * FO, TB), TB, 0, stream>>>(out, (size_t)N * FO);

  // 2) degrees -> dinv
  k_degree<<<blocks((size_t)ET, TB), TB, 0, stream>>>(ei, ew, deg, E, N);
  k_dinv<<<blocks((size_t)N, TB), TB, 0, stream>>>(deg, N);

  // 3) layer 1: h1pre = x @ W1  (WMMA f32)
  {
    const int tiles = (N / 16) * (FH / 16);
    k_gemm_wmma_f32<<<(tiles + 3) / 4, 128, 0, stream>>>(x, W1, h1pre, N, FH, FI);
  }
  // 4) aggregate with symmetric normalization, then bias + relu
  k_aggregate<<<ET, FH, 0, stream>>>(ei, ew, deg, h1pre, h1, E, FH);
  k_bias_relu<<<blocks((size_t)N * FH, TB), TB, 0, stream>>>(h1, b1, (size_t)N * FH, FH - 1);

  // 5) layer 2: h2pre = h1 @ W2  (WMMA f32)
  {
    const int tiles = (N / 16) * (FO / 16);
    k_gemm_wmma_f32<<<(tiles + 3) / 4, 128, 0, stream>>>(h1, W2, h2pre, N, FO, FH);
  }
  // 6) aggregate layer 2 directly into d_out
  k_aggregate<<<ET, FO, 0, stream>>>(ei, ew, deg, h2pre, out, E, FO);

  // 7) bias + row softmax (one wave per row, 64 cols)
  k_bias_softmax64<<<(N + 3) / 4, 128, 0, stream>>>(out, b2, N);
}